// GCN_74234214744298
// MI455X (gfx1250) — compile-verified
//
#include <hip/hip_runtime.h>
#include <hip/hip_bf16.h>

// ---------------------------------------------------------------------------
// GCN (3-layer GraphConv) for MI455X / gfx1250, wave32.
//   h1 = relu( (Din^-1/2 A Dout^-1/2 feat) W0 )
//   h2 = relu( (Din^-1/2 A Dout^-1/2 h1  ) W1 )
//   out =      (Din^-1/2 A Dout^-1/2 h2  ) W2 + b2
// Scatter-add is the bandwidth hog (L2-resident f32 atomics); GEMMs use
// V_WMMA_F32_16X16X4_F32 to keep full fp32 precision.
// ---------------------------------------------------------------------------

#define KDIM 128  // feature dim of all GEMM K-reductions

typedef __attribute__((ext_vector_type(2))) float v2f;
typedef __attribute__((ext_vector_type(8))) float v8f;

// ---- degree accumulation: deg_src / deg_dst via f32 atomics -----------------
__global__ void gcn_degree_kernel(const int* __restrict__ src,
                                  const int* __restrict__ dst,
                                  float* __restrict__ degS,
                                  float* __restrict__ degD, int n_edges) {
  int e = blockIdx.x * blockDim.x + threadIdx.x;
  if (e < n_edges) {
    unsafeAtomicAdd(&degS[src[e]], 1.0f);
    unsafeAtomicAdd(&degD[dst[e]], 1.0f);
  }
}

// ---- in-place deg -> rsqrt(max(deg,1))  (runs over normS||normD jointly) ---
__global__ void gcn_norm_kernel(float* __restrict__ d, int n) {
  int i = blockIdx.x * blockDim.x + threadIdx.x;
  if (i < n) d[i] = rsqrtf(fmaxf(d[i], 1.0f));
}

// ---- edge scatter: agg[dst] += h[src] * out_norm[src] ----------------------
// One wave32 per edge: lane l moves floats [4l,4l+3] of the 128-float row.
// src/dst are wave-uniform -> scalar loads; row read is a coalesced 512B.
__global__ void gcn_scatter_kernel(const float* __restrict__ h,
                                   const float* __restrict__ out_norm,
                                   const int* __restrict__ src,
                                   const int* __restrict__ dst,
                                   float* __restrict__ agg, int n_edges) {
  int wave = (int)((blockIdx.x * blockDim.x + threadIdx.x) >> 5);
  int lane = threadIdx.x & 31;
  if (wave >= n_edges) return;
  int s = src[wave];
  int d = dst[wave];
  float scale = out_norm[s];
  const float4* hp = (const float4*)(h + (size_t)s * KDIM);
  float4 v = hp[lane];
  float* ap = agg + (size_t)d * KDIM + lane * 4;
  unsafeAtomicAdd(ap + 0, v.x * scale);
  unsafeAtomicAdd(ap + 1, v.y * scale);
  unsafeAtomicAdd(ap + 2, v.z * scale);
  unsafeAtomicAdd(ap + 3, v.w * scale);
}

// ---- WMMA GEMM: out[M,N] = act( (agg * in_norm[:,None]) @ W  (+bias) ) -----
// Grid.x = ceil(M/16) (one 16-row tile per block). One wave per 16-col N-tile,
// blockDim.x = 32 * ceil(N/16). K = 128, stepped by 4 with f32 WMMA.
//
// Fragment layouts per ISA 7.12.2 (wave32):
//   A 16x4 f32 : lanes 0-15 hold M=lane, K={kk,kk+1}; lanes 16-31 K={kk+2,kk+3}
//   B 4x16 f32 : lanes 0-15 hold N=lane, K={kk,kk+1}; lanes 16-31 K={kk+2,kk+3}
//   C/D 16x16  : VGPR i -> (M = i + 8*(lane>=16), N = lane&15)
//
// OOB rows/cols are handled by CLAMPING the load address: a lane whose column
// is out of range accumulates garbage that lives only in that lane's C
// registers, and those are never stored. This keeps EXEC full (WMMA
// requirement) and keeps the inner loop branch-free.
__global__ void gcn_gemm_wmma_kernel(const float* __restrict__ A,
                                     const float* __restrict__ in_norm,
                                     const float* __restrict__ W,
                                     const float* __restrict__ bias,
                                     float* __restrict__ out,
                                     int M, int N, int do_relu) {
  const int m0   = blockIdx.x * 16;
  const int wv   = threadIdx.x >> 5;   // N-tile handled by this wave
  const int lane = threadIdx.x & 31;
  const int half = lane >> 4;          // 0: lanes 0-15, 1: lanes 16-31
  const int l15  = lane & 15;

  const int col    = wv * 16 + l15;
  const bool colOK = (col < N);
  const int colC   = colOK ? col : (N - 1);   // clamped load column

  int arow = m0 + l15;
  if (arow >= M) arow = M - 1;                // clamped load row
  const float ascale = in_norm[arow];

  // Per-lane streaming pointers; strength-reduced (no per-iter 64-bit muls).
  const float* Ap = A + (size_t)arow * KDIM + half * 2;
  const float* Wp = W + (size_t)(half * 2) * N + colC;
  const size_t wstep = (size_t)4 * N;

  v8f c = {};
#pragma unroll 4
  for (int kk = 0; kk < KDIM; kk += 4) {
    v2f a = *(const v2f*)Ap;           // K and K+1, contiguous, 8B aligned
    a *= ascale;                       // fuse D_in^{-1/2} scaling
    v2f b;
    b.x = Wp[0];                       // W[ka  ][col]
    b.y = Wp[N];                       // W[ka+1][col]
    // D = A x B + C  (fp32 matrix pipe)
    c = __builtin_amdgcn_wmma_f32_16x16x4_f32(
        /*neg_a=*/false, a, /*neg_b=*/false, b,
        /*c_mod=*/(short)0, c, /*reuse_a=*/false, /*reuse_b=*/false);
    Ap += 4;
    Wp += wstep;
  }

  if (colOK) {
    const float bv = bias ? bias[col] : 0.0f;
#pragma unroll
    for (int i = 0; i < 8; ++i) {
      const int row = m0 + i + half * 8;
      if (row < M) {
        float v = c[i] + bv;
        if (do_relu) v = fmaxf(v, 0.0f);
        out[(size_t)row * N + col] = v;
      }
    }
  }
}

// ---------------------------------------------------------------------------
extern "C" void kernel_launch(void* const* d_in, const int* in_sizes, int n_in,
                              void* d_out, int out_size, void* d_ws, size_t ws_size,
                              hipStream_t stream) {
  const float* feat = (const float*)d_in[0];
  const float* W0   = (const float*)d_in[1];
  const float* W1   = (const float*)d_in[2];
  const float* W2   = (const float*)d_in[3];
  const float* b2   = (const float*)d_in[4];
  const int*   src  = (const int*)d_in[5];
  const int*   dst  = (const int*)d_in[6];
  float* out = (float*)d_out;

  const int n_nodes = in_sizes[0] / KDIM;  // 100000
  const int n_out   = in_sizes[4];         // 40
  const int n_edges = in_sizes[5];         // 625000

  // ---- workspace layout ----
  char* ws = (char*)d_ws;
  float* normS = (float*)ws;               // deg_src -> out_norm (in place)
  float* normD = normS + n_nodes;          // deg_dst -> in_norm  (in place)
  size_t off = (((size_t)2 * n_nodes * sizeof(float)) + 511) & ~(size_t)511;
  float* agg  = (float*)(ws + off);        // 100000 x 128 f32 (51.2 MB)
  off += (size_t)n_nodes * KDIM * sizeof(float);
  float* hbuf = (float*)(ws + off);        // 100000 x 128 f32 (51.2 MB)

  const size_t featBytes = (size_t)n_nodes * KDIM * sizeof(float);
  const int mTiles = (n_nodes + 15) / 16;                      // 6250
  const int scatterBlocks = (n_edges + 7) / 8;                 // 8 waves/block
  const dim3 blk256(256);

  // ---- degree norms (once) ----
  hipMemsetAsync(normS, 0, (size_t)2 * n_nodes * sizeof(float), stream);
  gcn_degree_kernel<<<(n_edges + 255) / 256, blk256, 0, stream>>>(
      src, dst, normS, normD, n_edges);
  gcn_norm_kernel<<<(2 * n_nodes + 255) / 256, blk256, 0, stream>>>(
      normS, 2 * n_nodes);

  // ---- layer 0: feat -> hbuf ----
  hipMemsetAsync(agg, 0, featBytes, stream);
  gcn_scatter_kernel<<<scatterBlocks, blk256, 0, stream>>>(
      feat, normS, src, dst, agg, n_edges);
  gcn_gemm_wmma_kernel<<<mTiles, dim3(32 * 8), 0, stream>>>(
      agg, normD, W0, nullptr, hbuf, n_nodes, 128, /*relu=*/1);

  // ---- layer 1: hbuf -> hbuf ----
  hipMemsetAsync(agg, 0, featBytes, stream);
  gcn_scatter_kernel<<<scatterBlocks, blk256, 0, stream>>>(
      hbuf, normS, src, dst, agg, n_edges);
  gcn_gemm_wmma_kernel<<<mTiles, dim3(32 * 8), 0, stream>>>(
      agg, normD, W1, nullptr, hbuf, n_nodes, 128, /*relu=*/1);

  // ---- layer 2: hbuf -> out (N=40 -> 3 N-tiles -> 96 threads) ----
  hipMemsetAsync(agg, 0, featBytes, stream);
  gcn_scatter_kernel<<<scatterBlocks, blk256, 0, stream>>>(
      hbuf, normS, src, dst, agg, n_edges);
  gcn_gemm_wmma_kernel<<<mTiles, dim3(32 * ((n_out + 15) / 16)), 0, stream>>>(
      agg, normD, W2, b2, out, n_nodes, n_out, /*relu=*/0);
}